// Conv2D3_72026601554290
// MI455X (gfx1250) — compile-verified
//
#include <hip/hip_runtime.h>
#include <hip/hip_bf16.h>

// MI455X fused kernel:
//   phase 1: 7x7 VALID conv [4,3,70,70] -> H[64,128] tile in LDS (per block)
//   phase 2: 200x  H = W2*H + b2  via V_WMMA_F32_16X16X4_F32, H stays in LDS
//   phase 3: write tile to d_out in NCHW order
//
// fp32 WMMA chosen deliberately: 200 chained GEMMs amplify input quantization
// error ~ ||W2||^200; f16/bf16 operands would diverge from the fp32 reference.

typedef __attribute__((ext_vector_type(2))) float v2f;
typedef __attribute__((ext_vector_type(8))) float v8f;

#define NS       128          // spatial columns per workgroup tile
#define HSTRIDE  132          // padded LDS row stride for H (floats)
#define W2STRIDE 65           // padded LDS row stride for W2 (floats)
#define NREPEAT  200

__global__ __launch_bounds__(256, 1)
void conv_chain_wmma_kernel(const float* __restrict__ x,    // [4,3,70,70]
                            const float* __restrict__ W1,   // [64,3,7,7]
                            const float* __restrict__ b1,   // [64]
                            const float* __restrict__ W2,   // [64,64] (O,I)
                            const float* __restrict__ b2,   // [64]
                            float* __restrict__ out)        // [4,64,64,64]
{
    __shared__ float Hbuf[64 * HSTRIDE];     // 33.8 KB, h tile [c][s]
    __shared__ float W2lds[64 * W2STRIDE];   // 16.6 KB
    __shared__ float b2lds[64];

    const int tid  = threadIdx.x;
    const int lane = tid & 31;
    const int wave = tid >> 5;        // 0..7 -> spatial sub-tile
    const int half = lane >> 4;       // 0: lanes 0-15, 1: lanes 16-31
    const int m    = lane & 15;       // row (A) / column (B,C,D) within tile

    // ---- stage W2, b2 into LDS (padded) ----
    for (int idx = tid; idx < 64 * 64; idx += 256)
        W2lds[(idx >> 6) * W2STRIDE + (idx & 63)] = W2[idx];
    if (tid < 64) b2lds[tid] = b2[tid];

    // ---- phase 1: direct 7x7 conv into Hbuf[c][s] ----
    for (int idx = tid; idx < 64 * NS; idx += 256) {
        const int c0  = idx / NS;
        const int s   = idx - c0 * NS;
        const int sg  = blockIdx.x * NS + s;           // global spatial index
        const int b   = sg >> 12;                      // batch
        const int rem = sg & 4095;
        const int yy  = rem >> 6;
        const int xx  = rem & 63;
        float acc = b1[c0];
        for (int ci = 0; ci < 3; ++ci) {
            const float* xp = x  + ((b * 3 + ci) * 70 + yy) * 70 + xx;
            const float* wp = W1 + (c0 * 3 + ci) * 49;
            #pragma unroll
            for (int ky = 0; ky < 7; ++ky) {
                #pragma unroll
                for (int kx = 0; kx < 7; ++kx)
                    acc = __builtin_fmaf(xp[ky * 70 + kx], wp[ky * 7 + kx], acc);
            }
        }
        Hbuf[c0 * HSTRIDE + s] = acc;
    }
    __syncthreads();

    // ---- hoist loop-invariant A fragments (W2) into VGPRs ----
    // ISA 7.12.2, 32-bit A 16x4: lanes 0-15 hold K=0,1; lanes 16-31 hold K=2,3
    v2f afrag[4][16];
    #pragma unroll
    for (int ct = 0; ct < 4; ++ct) {
        #pragma unroll
        for (int k = 0; k < 16; ++k) {
            const int row = ct * 16 + m;
            const int c   = k * 4 + half * 2;
            v2f a;
            a[0] = W2lds[row * W2STRIDE + c];
            a[1] = W2lds[row * W2STRIDE + c + 1];
            afrag[ct][k] = a;
        }
    }

    // ---- bias as initial C fragment (C/D: vgpr i = rows i and i+8) ----
    v8f biasv[4];
    #pragma unroll
    for (int ct = 0; ct < 4; ++ct) {
        v8f bv;
        #pragma unroll
        for (int i = 0; i < 8; ++i)
            bv[i] = b2lds[ct * 16 + i + half * 8];
        biasv[ct] = bv;
    }

    const int col = wave * 16 + m;   // this lane's spatial column

    // ---- phase 2: 200 chained channel GEMMs, H resident in LDS ----
    for (int it = 0; it < NREPEAT; ++it) {
        v8f acc0 = biasv[0], acc1 = biasv[1], acc2 = biasv[2], acc3 = biasv[3];
        #pragma unroll
        for (int k = 0; k < 16; ++k) {
            // B 4x16 fragment: vgpr j = rows K=j (lanes 0-15) and K=j+2 (16-31)
            const int r = k * 4 + half * 2;
            v2f bfrag;
            bfrag[0] = Hbuf[r * HSTRIDE + col];
            bfrag[1] = Hbuf[(r + 1) * HSTRIDE + col];
            acc0 = __builtin_amdgcn_wmma_f32_16x16x4_f32(false, afrag[0][k],
                       false, bfrag, (short)0, acc0, false, false);
            acc1 = __builtin_amdgcn_wmma_f32_16x16x4_f32(false, afrag[1][k],
                       false, bfrag, (short)0, acc1, false, false);
            acc2 = __builtin_amdgcn_wmma_f32_16x16x4_f32(false, afrag[2][k],
                       false, bfrag, (short)0, acc2, false, false);
            acc3 = __builtin_amdgcn_wmma_f32_16x16x4_f32(false, afrag[3][k],
                       false, bfrag, (short)0, acc3, false, false);
        }
        __syncthreads();   // all reads of Hbuf done
        #pragma unroll
        for (int i = 0; i < 8; ++i) {
            const int row = i + half * 8;
            Hbuf[(row)      * HSTRIDE + col] = acc0[i];
            Hbuf[(row + 16) * HSTRIDE + col] = acc1[i];
            Hbuf[(row + 32) * HSTRIDE + col] = acc2[i];
            Hbuf[(row + 48) * HSTRIDE + col] = acc3[i];
        }
        __syncthreads();   // writes visible before next iteration
    }

    // ---- phase 3: write tile out in NCHW ----
    for (int idx = tid; idx < 64 * NS; idx += 256) {
        const int c0  = idx / NS;
        const int s   = idx - c0 * NS;
        const int sg  = blockIdx.x * NS + s;
        const int b   = sg >> 12;
        const int rem = sg & 4095;
        out[((b * 64 + c0) << 12) + rem] = Hbuf[c0 * HSTRIDE + s];
    }
}

extern "C" void kernel_launch(void* const* d_in, const int* in_sizes, int n_in,
                              void* d_out, int out_size, void* d_ws, size_t ws_size,
                              hipStream_t stream) {
    const float* x  = (const float*)d_in[0];
    const float* W1 = (const float*)d_in[1];
    const float* b1 = (const float*)d_in[2];
    const float* W2 = (const float*)d_in[3];
    const float* b2 = (const float*)d_in[4];
    float* out = (float*)d_out;

    // 4*64*64 = 16384 spatial positions / NS per block
    dim3 grid(16384 / NS), block(256);
    conv_chain_wmma_kernel<<<grid, block, 0, stream>>>(x, W1, b1, W2, b2, out);
}